// MaterialPointModel2d_38285338477002
// MI455X (gfx1250) — compile-verified
//
#include <hip/hip_runtime.h>
#include <math.h>

// ---------------- constants (match reference) ----------------
#define QUALITY 8
constexpr int   N_GRID = 128 * QUALITY;            // 1024
constexpr int   G2     = N_GRID * N_GRID;          // 1,048,576 nodes
constexpr float DXC    = 1.0f / (float)N_GRID;
constexpr float INV_DX = (float)N_GRID;
constexpr float DT     = 1e-4f / (float)QUALITY;   // 1.25e-5
constexpr float P_VOL  = (DXC * 0.5f) * (DXC * 0.5f);
constexpr float P_MASS = P_VOL * 1.0f;
constexpr float E_MOD  = 5000.0f;
constexpr float NU     = 0.2f;
constexpr float MU0    = E_MOD / (2.0f * (1.0f + NU));
constexpr float LAM0   = E_MOD * NU / ((1.0f + NU) * (1.0f - 2.0f * NU));

__device__ __forceinline__ void atomic_add_f32(float* p, float v) {
    // relaxed + agent scope -> expect global_atomic_add_f32 (no return) on gfx1250
    __hip_atomic_fetch_add(p, v, __ATOMIC_RELAXED, __HIP_MEMORY_SCOPE_AGENT);
}

__device__ __forceinline__ float clampf(float v, float lo, float hi) {
    return fminf(fmaxf(v, lo), hi);
}

// ---------------- kernel 0: atomic lowering probe (never launched) ----------
// Placed first in the file so the disasm snippet shows how the fp32 atomic
// lowers on gfx1250 (hardware global_atomic_add_f32 vs CAS loop).
__global__ void probe_atomic(float* p, float v) {
    atomic_add_f32(p + threadIdx.x, v);
}

// ---------------- kernel 1: clear grid scratch ----------------
__global__ __launch_bounds__(256) void zero_grid(float4* __restrict__ ws, int n4) {
    int i = blockIdx.x * blockDim.x + threadIdx.x;
    if (i < n4) ws[i] = make_float4(0.f, 0.f, 0.f, 0.f);
}

// Grid scratch layout: float4 per cell = {mom_x, mom_y, mass, pad}
// -> the 3 atomics per (particle, offset) hit one 16B sector, not 3 lines.

// ---------------- kernel 2: P2G (+ F / Jp update) ----------------
__global__ __launch_bounds__(256) void p2g(
    const float2* __restrict__ x, const float2* __restrict__ v,
    const float4* __restrict__ C, const float4* __restrict__ F,
    const float*  __restrict__ Jp, const int* __restrict__ mat,
    float* __restrict__ grid,                 // float4 per cell (AoS)
    float4* __restrict__ outF, float* __restrict__ outJp, int nP)
{
    int p = blockIdx.x * blockDim.x + threadIdx.x;
    if (p >= nP) return;

    // stream ahead: next wavefront-batch of particle records (global_prefetch_b8)
    if (p + 2048 < nP) {
        __builtin_prefetch(&x[p + 2048], 0, 1);
        __builtin_prefetch(&C[p + 2048], 0, 1);
        __builtin_prefetch(&F[p + 2048], 0, 1);
    }

    float2 xp = x[p];
    float bxf = floorf(xp.x * INV_DX - 0.5f);
    float byf = floorf(xp.y * INV_DX - 0.5f);
    int   bx  = (int)bxf, by = (int)byf;
    float fxx = xp.x * INV_DX - bxf;
    float fxy = xp.y * INV_DX - byf;

    float wx[3], wy[3];
    wx[0] = 0.5f * (1.5f - fxx) * (1.5f - fxx);
    wx[1] = 0.75f - (fxx - 1.0f) * (fxx - 1.0f);
    wx[2] = 0.5f * (fxx - 0.5f) * (fxx - 0.5f);
    wy[0] = 0.5f * (1.5f - fxy) * (1.5f - fxy);
    wy[1] = 0.75f - (fxy - 1.0f) * (fxy - 1.0f);
    wy[2] = 0.5f * (fxy - 0.5f) * (fxy - 0.5f);

    float4 Cm = C[p];                       // row-major [[x,y],[z,w]]
    float4 Fm = F[p];
    // F = (I + DT*C) * F
    float a00 = 1.f + DT * Cm.x, a01 = DT * Cm.y;
    float a10 = DT * Cm.z,       a11 = 1.f + DT * Cm.w;
    float f00 = a00 * Fm.x + a01 * Fm.z;
    float f01 = a00 * Fm.y + a01 * Fm.w;
    float f10 = a10 * Fm.x + a11 * Fm.z;
    float f11 = a10 * Fm.y + a11 * Fm.w;

    float jp = Jp[p];
    int   m  = mat[p];

    float h = clampf(expf(10.0f * (1.0f - jp)), 0.1f, 5.0f);
    if (m == 1) h = 0.3f;
    float mu = (m == 0) ? 0.0f : MU0 * h;
    float la = LAM0 * h;

    // ---- closed-form 2x2 SVD: F = U * diag(s0,s1) * Vh ----
    float Ee = 0.5f * (f00 + f11), Ff = 0.5f * (f00 - f11);
    float Gg = 0.5f * (f10 + f01), Hh = 0.5f * (f10 - f01);
    float Q  = sqrtf(Ee * Ee + Hh * Hh);
    float Rr = sqrtf(Ff * Ff + Gg * Gg);
    float s0 = Q + Rr;
    float s1 = Q - Rr;
    float a1 = atan2f(Gg, Ff);
    float a2 = atan2f(Hh, Ee);
    float beta = 0.5f * (a2 - a1);          // Vh = R(beta)
    float gam  = 0.5f * (a2 + a1);          // U  = R(gam)
    float su, cu, sv, cv;
    __sincosf(gam,  &su, &cu);
    __sincosf(beta, &sv, &cv);
    float sgn = (s1 < 0.f) ? -1.f : 1.f;    // fold sign into 2nd row of Vh
    s1 = fabsf(s1);
    float vh00 = cv,        vh01 = -sv;
    float vh10 = sgn * sv,  vh11 = sgn * cv;

    // plastic clamp (snow)
    float ns0 = s0, ns1 = s1;
    if (m == 2) {
        ns0 = clampf(s0, 1.0f - 2.5e-2f, 1.0f + 4.5e-3f);
        ns1 = clampf(s1, 1.0f - 2.5e-2f, 1.0f + 4.5e-3f);
    }
    jp = jp * (s0 / ns0) * (s1 / ns1);
    float J = ns0 * ns1;

    if (m == 0) {                            // fluid: reset to sqrt(J)*I
        float sj = sqrtf(J);
        f00 = sj; f01 = 0.f; f10 = 0.f; f11 = sj;
    } else if (m == 2) {                     // snow: U*diag(ns)*Vh
        float u0 = cu * ns0, u1 = -su * ns1;
        float u2 = su * ns0, u3 =  cu * ns1;
        f00 = u0 * vh00 + u1 * vh10;  f01 = u0 * vh01 + u1 * vh11;
        f10 = u2 * vh00 + u3 * vh10;  f11 = u2 * vh01 + u3 * vh11;
    }

    // R = U * Vh
    float r00 = cu * vh00 - su * vh10;  float r01 = cu * vh01 - su * vh11;
    float r10 = su * vh00 + cu * vh10;  float r11 = su * vh01 + cu * vh11;

    // stress = 2*mu*(F-R)*F^T + la*J*(J-1)*I, then scaled
    float d00 = f00 - r00, d01 = f01 - r01, d10 = f10 - r10, d11 = f11 - r11;
    float lj  = la * J * (J - 1.0f);
    float st00 = 2.f * mu * (d00 * f00 + d01 * f01) + lj;
    float st01 = 2.f * mu * (d00 * f10 + d01 * f11);
    float st10 = 2.f * mu * (d10 * f00 + d11 * f01);
    float st11 = 2.f * mu * (d10 * f10 + d11 * f11) + lj;
    float k = -DT * P_VOL * 4.0f * INV_DX * INV_DX;
    float af00 = k * st00 + P_MASS * Cm.x;
    float af01 = k * st01 + P_MASS * Cm.y;
    float af10 = k * st10 + P_MASS * Cm.z;
    float af11 = k * st11 + P_MASS * Cm.w;

    outF[p]  = make_float4(f00, f01, f10, f11);
    outJp[p] = jp;

    float2 vp  = v[p];
    float  mvx = P_MASS * vp.x, mvy = P_MASS * vp.y;
    float  wm  = P_MASS;

    #pragma unroll
    for (int i = 0; i < 3; ++i) {
        #pragma unroll
        for (int j = 0; j < 3; ++j) {
            float wgt = wx[i] * wy[j];
            float dpx = ((float)i - fxx) * DXC;
            float dpy = ((float)j - fxy) * DXC;
            float momx = wgt * (mvx + af00 * dpx + af01 * dpy);
            float momy = wgt * (mvy + af10 * dpx + af11 * dpy);
            int gi = min(max(bx + i, 0), N_GRID - 1);
            int gj = min(max(by + j, 0), N_GRID - 1);
            float* cell = grid + 4 * (size_t)(gi * N_GRID + gj);
            atomic_add_f32(cell + 0, momx);
            atomic_add_f32(cell + 1, momy);
            atomic_add_f32(cell + 2, wgt * wm);
        }
    }
}

// ---------------- kernel 3: grid update ----------------
__global__ __launch_bounds__(256) void grid_update(
    float4* __restrict__ grid,
    const float* __restrict__ grav, const float* __restrict__ astr,
    const float* __restrict__ apos)
{
    int idx = blockIdx.x * blockDim.x + threadIdx.x;
    if (idx >= G2) return;
    int ii = idx >> 10;                 // idx / N_GRID
    int jj = idx & (N_GRID - 1);        // idx % N_GRID
    float4 cell = grid[idx];            // one global_load_b128 per node
    float vx = 0.f, vy = 0.f;
    if (cell.z > 0.f) {
        float inv = 1.0f / cell.z;
        vx = cell.x * inv;
        vy = cell.y * inv;
        vx += DT * 30.0f * grav[0];
        vy += DT * 30.0f * grav[1];
        float dx_ = apos[0] - DXC * (float)ii;
        float dy_ = apos[1] - DXC * (float)jj;
        float nrm = sqrtf(dx_ * dx_ + dy_ * dy_);
        float sc  = astr[0] * DT * 100.0f / (0.01f + nrm);
        vx += dx_ * sc;
        vy += dy_ * sc;
        if (ii < 3 && vx < 0.f)          vx = 0.f;
        if (ii > N_GRID - 3 && vx > 0.f) vx = 0.f;
        if (jj < 3 && vy < 0.f)          vy = 0.f;
        if (jj > N_GRID - 3 && vy > 0.f) vy = 0.f;
    }
    // write back velocity as one b64 store (first two floats of the cell)
    *(float2*)&grid[idx] = make_float2(vx, vy);
}

// ---------------- kernel 4: G2P ----------------
__global__ __launch_bounds__(256) void g2p(
    const float2* __restrict__ x,
    const float4* __restrict__ grid,
    float2* __restrict__ out_x, float2* __restrict__ out_v,
    float4* __restrict__ out_C, int nP)
{
    int p = blockIdx.x * blockDim.x + threadIdx.x;
    if (p >= nP) return;

    if (p + 2048 < nP) __builtin_prefetch(&x[p + 2048], 0, 1);

    float2 xp = x[p];
    float bxf = floorf(xp.x * INV_DX - 0.5f);
    float byf = floorf(xp.y * INV_DX - 0.5f);
    int   bx  = (int)bxf, by = (int)byf;
    float fxx = xp.x * INV_DX - bxf;
    float fxy = xp.y * INV_DX - byf;

    float wx[3], wy[3];
    wx[0] = 0.5f * (1.5f - fxx) * (1.5f - fxx);
    wx[1] = 0.75f - (fxx - 1.0f) * (fxx - 1.0f);
    wx[2] = 0.5f * (fxx - 0.5f) * (fxx - 0.5f);
    wy[0] = 0.5f * (1.5f - fxy) * (1.5f - fxy);
    wy[1] = 0.75f - (fxy - 1.0f) * (fxy - 1.0f);
    wy[2] = 0.5f * (fxy - 0.5f) * (fxy - 0.5f);

    float nvx = 0.f, nvy = 0.f;
    float c00 = 0.f, c01 = 0.f, c10 = 0.f, c11 = 0.f;

    #pragma unroll
    for (int i = 0; i < 3; ++i) {
        #pragma unroll
        for (int j = 0; j < 3; ++j) {
            float wgt = wx[i] * wy[j];
            int gi = min(max(bx + i, 0), N_GRID - 1);
            int gj = min(max(by + j, 0), N_GRID - 1);
            // one global_load_b64 per node: (vx, vy)
            float2 g = *(const float2*)&grid[gi * N_GRID + gj];
            nvx += wgt * g.x;
            nvy += wgt * g.y;
            float dpx = (float)i - fxx;       // unscaled for C
            float dpy = (float)j - fxy;
            c00 += wgt * g.x * dpx;  c01 += wgt * g.x * dpy;
            c10 += wgt * g.y * dpx;  c11 += wgt * g.y * dpy;
        }
    }
    float s = 4.0f * INV_DX;
    out_v[p] = make_float2(nvx, nvy);
    out_C[p] = make_float4(s * c00, s * c01, s * c10, s * c11);
    out_x[p] = make_float2(xp.x + DT * nvx, xp.y + DT * nvy);
}

// ---------------- host launcher ----------------
extern "C" void kernel_launch(void* const* d_in, const int* in_sizes, int n_in,
                              void* d_out, int out_size, void* d_ws, size_t ws_size,
                              hipStream_t stream) {
    const float2* x    = (const float2*)d_in[0];
    const float2* v    = (const float2*)d_in[1];
    const float4* C    = (const float4*)d_in[2];
    const float4* F    = (const float4*)d_in[3];
    const float*  Jp   = (const float*) d_in[4];
    const float*  grav = (const float*) d_in[5];
    const float*  astr = (const float*) d_in[6];
    const float*  apos = (const float*) d_in[7];
    const int*    mat  = (const int*)  d_in[8];

    const int nP = in_sizes[0] / 2;

    // workspace: float4 per grid cell = {mom_x, mom_y, mass, pad}, 16 MB
    float4* grid = (float4*)d_ws;

    // output layout (tuple order): x(2P) | v(2P) | C(4P) | F(4P) | Jp(P)
    float*  out    = (float*)d_out;
    float2* out_x  = (float2*)(out);
    float2* out_v  = (float2*)(out + 2 * (size_t)nP);
    float4* out_C  = (float4*)(out + 4 * (size_t)nP);
    float4* out_F  = (float4*)(out + 8 * (size_t)nP);
    float*  out_Jp = out + 12 * (size_t)nP;

    const int BLK = 256;                      // 8 wave32 waves per block
    zero_grid<<<(G2 + BLK - 1) / BLK, BLK, 0, stream>>>(grid, G2);

    p2g<<<(nP + BLK - 1) / BLK, BLK, 0, stream>>>(
        x, v, C, F, Jp, mat, (float*)grid, out_F, out_Jp, nP);

    grid_update<<<(G2 + BLK - 1) / BLK, BLK, 0, stream>>>(
        grid, grav, astr, apos);

    g2p<<<(nP + BLK - 1) / BLK, BLK, 0, stream>>>(
        x, grid, out_x, out_v, out_C, nP);
}